// YoloxLoss_31310311588008
// MI455X (gfx1250) — compile-verified
//
#include <hip/hip_runtime.h>
#include <hip/hip_bf16.h>

#define Bn 32
#define An 8400
#define Cn 80
#define Gn 50
#define BA (Bn * An)           // 268800
#define BG (Bn * Gn)           // 1600
#define NPART 1050             // BA / 256

typedef float v2f __attribute__((ext_vector_type(2)));
typedef float v8f __attribute__((ext_vector_type(8)));

__device__ __forceinline__ float softplusf(float x) {
    return fmaxf(x, 0.f) + log1pf(expf(-fabsf(x)));
}
__device__ __forceinline__ float logsigf(float x) {
    return fminf(x, 0.f) - log1pf(expf(-fabsf(x)));
}
__device__ __forceinline__ float qprob(float cl, float lo) {
    float lq = 0.5f * (logsigf(cl) + lo);
    float q = expf(lq);
    return fminf(fmaxf(q, 1e-7f), 1.f - 1e-7f);
}
__device__ __forceinline__ float iou_cxcywh(float ax, float ay, float aw, float ah,
                                            float bx, float by, float bw, float bh) {
    float tlx = fmaxf(ax - aw * 0.5f, bx - bw * 0.5f);
    float tly = fmaxf(ay - ah * 0.5f, by - bh * 0.5f);
    float brx = fminf(ax + aw * 0.5f, bx + bw * 0.5f);
    float bry = fminf(ay + ah * 0.5f, by + bh * 0.5f);
    float en = (tlx < brx && tly < bry) ? 1.f : 0.f;
    float ai = (brx - tlx) * (bry - tly) * en;
    return ai / (aw * ah + bw * bh - ai + 1e-16f);
}
__device__ __forceinline__ void pair_geom(float xc, float yc, float sg,
                                          float gx, float gy, float gw, float gh,
                                          bool valid, bool& in_box, bool& in_ctr) {
    float l = xc - (gx - gw * 0.5f);
    float r = (gx + gw * 0.5f) - xc;
    float tt = yc - (gy - gh * 0.5f);
    float bb = (gy + gh * 0.5f) - yc;
    in_box = (fminf(fminf(l, r), fminf(tt, bb)) > 0.f) && valid;
    float cr = 2.5f * sg;
    float l2 = xc - (gx - cr);
    float r2 = (gx + cr) - xc;
    float t2 = yc - (gy - cr);
    float b2 = (gy + cr) - yc;
    in_ctr = (fminf(fminf(l2, r2), fminf(t2, b2)) > 0.f) && valid;
}

// ---------------------------------------------------------------------------
// Kernel 1: s_neg[b,a] = sum_c log1p(-q) via V_WMMA_F32_16X16X4_F32 row-sums.
// Each wave32 owns 16 anchors (rows M). A-matrix layout per ISA: lanes 0-15
// hold (K0,K1), lanes 16-31 hold (K2,K3). B = all-ones -> D[m][n] = row sum.
// ---------------------------------------------------------------------------
__global__ __launch_bounds__(256) void k_sneg(const float* __restrict__ cls,
                                              const float* __restrict__ obj,
                                              float* __restrict__ sneg) {
    int lane = threadIdx.x & 31;
    int wave = threadIdx.x >> 5;
    long base = (long)blockIdx.x * 128 + (long)wave * 16;   // flat b*A+a
    long i = base + (lane & 15);
    int khalf = lane >> 4;
    float lo = logsigf(obj[i]);
    const float* crow = cls + i * Cn;
    v8f acc = {};
    v2f bmat;
    bmat.x = 1.0f;
    bmat.y = 1.0f;
    for (int cb = 0; cb < Cn; cb += 4) {
        int c0 = cb + khalf * 2;
        v2f amat;
        amat.x = log1pf(-qprob(crow[c0], lo));
        amat.y = log1pf(-qprob(crow[c0 + 1], lo));
        acc = __builtin_amdgcn_wmma_f32_16x16x4_f32(false, amat, false, bmat,
                                                    (short)0, acc, false, false);
    }
    // C/D layout: VGPR r -> row r (lanes 0-15, N=lane) / row 8+r (lanes 16-31).
    // Row sums replicated over N, so lanes with N==0 extract 8 rows each.
    if ((lane & 15) == 0) {
        long out = base + 8 * khalf;
        #pragma unroll
        for (int r = 0; r < 8; r++) sneg[out + r] = acc[r];
    }
}

// ---------------------------------------------------------------------------
// Kernel 2: fg0[b,a] = any_g (in_boxes | in_ctr)
// ---------------------------------------------------------------------------
__global__ __launch_bounds__(256) void k_fg0(const float* __restrict__ labels,
                                             const float* __restrict__ xs,
                                             const float* __restrict__ ys,
                                             const float* __restrict__ ss,
                                             float* __restrict__ fg0) {
    long i = (long)blockIdx.x * 256 + threadIdx.x;
    if (i >= (long)BA) return;
    int b = (int)(i / An);
    int a = (int)(i % An);
    float sg = ss[a];
    float xc = (xs[a] + 0.5f) * sg;
    float yc = (ys[a] + 0.5f) * sg;
    const float* lb = labels + (long)b * Gn * 5;
    bool any = false;
    for (int g = 0; g < Gn; g++) {
        const float* L = lb + g * 5;
        bool valid = (L[0] + L[1] + L[2] + L[3] + L[4]) > 0.f;
        bool ib, ic;
        pair_geom(xc, yc, sg, L[1], L[2], L[3], L[4], valid, ib, ic);
        any |= (ib || ic);
    }
    fg0[i] = any ? 1.f : 0.f;
}

// ---------------------------------------------------------------------------
// Kernel 3: per (b,g): dyn_k (sum of top-10 ious) and first dyn_k smallest
// costs (stable ties by index) via LDS-resident row + iterative selection.
// ---------------------------------------------------------------------------
__global__ __launch_bounds__(256) void k_rows(const float* __restrict__ pred,
                                              const float* __restrict__ obj,
                                              const float* __restrict__ cls,
                                              const float* __restrict__ labels,
                                              const float* __restrict__ xs,
                                              const float* __restrict__ ys,
                                              const float* __restrict__ ss,
                                              const float* __restrict__ sneg,
                                              const float* __restrict__ fg0,
                                              int* __restrict__ dynk,
                                              int* __restrict__ topk) {
    __shared__ float srow[An];
    __shared__ float rv[256];
    __shared__ int ri[256];
    int bg = blockIdx.x;
    int b = bg / Gn;
    int g = bg % Gn;
    int t = threadIdx.x;
    const float* L = labels + ((long)b * Gn + g) * 5;
    float l0 = L[0], gx = L[1], gy = L[2], gw = L[3], gh = L[4];
    bool valid = (l0 + gx + gy + gw + gh) > 0.f;
    if (!valid) {
        if (t == 0) dynk[bg] = 0;
        return;
    }
    int gcls = (int)l0;
    long abase = (long)b * An;

    // ---- Phase A: ious row, top-10 sum -> dyn_k ----
    for (int a = t; a < An; a += 256) {
        const float* P = pred + (abase + a) * 85;
        srow[a] = iou_cxcywh(gx, gy, gw, gh, P[0], P[1], P[2], P[3]) * fg0[abase + a];
    }
    __syncthreads();
    float topsum = 0.f;
    for (int it = 0; it < 10; it++) {
        float bv = -1.f;
        int bi = An;
        for (int a = t; a < An; a += 256) {
            float v = srow[a];
            if (v > bv || (v == bv && a < bi)) { bv = v; bi = a; }
        }
        rv[t] = bv; ri[t] = bi;
        __syncthreads();
        for (int st = 128; st > 0; st >>= 1) {
            if (t < st) {
                float ov = rv[t + st]; int oi = ri[t + st];
                if (ov > rv[t] || (ov == rv[t] && oi < ri[t])) { rv[t] = ov; ri[t] = oi; }
            }
            __syncthreads();
        }
        topsum += rv[0];
        if (t == 0 && ri[0] < An) srow[ri[0]] = -1.f;
        __syncthreads();
    }
    int dk = (int)topsum;
    if (dk < 1) dk = 1;
    if (dk > 10) dk = 10;
    if (t == 0) dynk[bg] = dk;

    // ---- Phase B: cost row, first-10 smallest (ties -> lowest index) ----
    for (int a = t; a < An; a += 256) {
        long ia = abase + a;
        const float* P = pred + ia * 85;
        float sg = ss[a];
        float xc = (xs[a] + 0.5f) * sg;
        float yc = (ys[a] + 0.5f) * sg;
        bool ib, ic;
        pair_geom(xc, yc, sg, gx, gy, gw, gh, true, ib, ic);
        float f0 = fg0[ia];
        float iou_m = iou_cxcywh(gx, gy, gw, gh, P[0], P[1], P[2], P[3]) * f0;
        float lo = logsigf(obj[ia]);
        float qg = qprob(cls[ia * Cn + gcls], lo);
        float cost = -logf(qg) + log1pf(-qg) - sneg[ia] - 3.f * logf(iou_m + 1e-8f);
        if (!(ib && ic)) cost += 1e5f;
        if (f0 == 0.f) cost += 1e9f;
        srow[a] = cost;
    }
    __syncthreads();
    for (int it = 0; it < 10; it++) {
        float bv = __builtin_inff();
        int bi = An;
        for (int a = t; a < An; a += 256) {
            float v = srow[a];
            if (v < bv || (v == bv && a < bi)) { bv = v; bi = a; }
        }
        rv[t] = bv; ri[t] = bi;
        __syncthreads();
        for (int st = 128; st > 0; st >>= 1) {
            if (t < st) {
                float ov = rv[t + st]; int oi = ri[t + st];
                if (ov < rv[t] || (ov == rv[t] && oi < ri[t])) { rv[t] = ov; ri[t] = oi; }
            }
            __syncthreads();
        }
        if (t == 0) {
            topk[bg * 10 + it] = ri[0];
            if (ri[0] < An) srow[ri[0]] = __builtin_inff();
        }
        __syncthreads();
    }
}

// ---------------------------------------------------------------------------
// Kernel 4: per anchor: matching count/mask, best_gt, multi-match resolution,
// fg / matched / pred_iou.
// ---------------------------------------------------------------------------
__global__ __launch_bounds__(256) void k_cols(const float* __restrict__ pred,
                                              const float* __restrict__ obj,
                                              const float* __restrict__ cls,
                                              const float* __restrict__ labels,
                                              const float* __restrict__ xs,
                                              const float* __restrict__ ys,
                                              const float* __restrict__ ss,
                                              const float* __restrict__ sneg,
                                              const float* __restrict__ fg0,
                                              const int* __restrict__ dynk,
                                              const int* __restrict__ topk,
                                              float* __restrict__ fg,
                                              int* __restrict__ matched,
                                              float* __restrict__ prediou) {
    long i = (long)blockIdx.x * 256 + threadIdx.x;
    if (i >= (long)BA) return;
    int b = (int)(i / An);
    int a = (int)(i % An);
    const float* P = pred + i * 85;
    float px = P[0], py = P[1], pw = P[2], ph = P[3];
    float sg = ss[a];
    float xc = (xs[a] + 0.5f) * sg;
    float yc = (ys[a] + 0.5f) * sg;
    float f0 = fg0[i];
    float lo = logsigf(obj[i]);
    float sn = sneg[i];
    const float* lb = labels + (long)b * Gn * 5;
    float bestc = __builtin_inff();
    int bestg = 0;
    unsigned long long mmask = 0ULL;
    int mcount = 0;
    for (int g = 0; g < Gn; g++) {
        const float* L = lb + g * 5;
        float l0 = L[0], gx = L[1], gy = L[2], gw = L[3], gh = L[4];
        bool valid = (l0 + gx + gy + gw + gh) > 0.f;
        bool ib, ic;
        pair_geom(xc, yc, sg, gx, gy, gw, gh, valid, ib, ic);
        float vf = valid ? 1.f : 0.f;
        float iou_m = iou_cxcywh(gx, gy, gw, gh, px, py, pw, ph) * f0 * vf;
        int gcls = (int)l0;
        float qg = qprob(cls[i * Cn + gcls], lo);
        float cost = -logf(qg) + log1pf(-qg) - sn - 3.f * logf(iou_m + 1e-8f);
        if (!(ib && ic)) cost += 1e5f;
        if (f0 == 0.f || !valid) cost += 1e9f;
        if (cost < bestc) { bestc = cost; bestg = g; }   // first index wins ties
        int bgidx = b * Gn + g;
        int dk = dynk[bgidx];
        bool m = false;
        for (int tt = 0; tt < dk; tt++)
            if (topk[bgidx * 10 + tt] == a) { m = true; break; }
        if (m) { mmask |= 1ULL << g; mcount++; }
    }
    int fgv, mg;
    if (mcount > 1)      { mg = bestg; fgv = 1; }
    else if (mcount == 1){ mg = __builtin_ctzll(mmask); fgv = 1; }
    else                 { mg = 0; fgv = 0; }
    float pi = 0.f;
    if (fgv) {
        const float* L = lb + mg * 5;
        float vf = ((L[0] + L[1] + L[2] + L[3] + L[4]) > 0.f) ? 1.f : 0.f;
        pi = iou_cxcywh(L[1], L[2], L[3], L[4], px, py, pw, ph) * f0 * vf;
    }
    fg[i] = (float)fgv;
    matched[i] = mg;
    prediou[i] = pi;
}

// ---------------------------------------------------------------------------
// Kernel 5: per-anchor loss terms, per-block tree reduction to partials.
// ---------------------------------------------------------------------------
__global__ __launch_bounds__(256) void k_loss(const float* __restrict__ pred,
                                              const float* __restrict__ obj,
                                              const float* __restrict__ cls,
                                              const float* __restrict__ labels,
                                              const float* __restrict__ fg,
                                              const int* __restrict__ matched,
                                              const float* __restrict__ prediou,
                                              float* __restrict__ partials) {
    __shared__ float s0[256], s1[256], s2[256], s3[256];
    long i = (long)blockIdx.x * 256 + threadIdx.x;
    float vfg = 0.f, viou = 0.f, vobj = 0.f, vcls = 0.f;
    if (i < (long)BA) {
        int b = (int)(i / An);
        const float* P = pred + i * 85;
        float o = obj[i];
        float f = fg[i];
        vfg = f;
        vobj = softplusf(o) - o * f;
        if (f > 0.f) {
            int mg = matched[i];
            const float* L = labels + ((long)b * Gn + mg) * 5;
            float iouv = iou_cxcywh(P[0], P[1], P[2], P[3], L[1], L[2], L[3], L[4]);
            viou = 1.f - iouv * iouv;
            int tc = (int)L[0];
            float pi = prediou[i];
            const float* cr = cls + i * Cn;
            float csum = 0.f;
            for (int c = 0; c < Cn; c++) csum += softplusf(cr[c]);
            csum -= cr[tc] * pi;   // f == 1 here
            vcls = csum;
        }
    }
    int t = threadIdx.x;
    s0[t] = vfg; s1[t] = viou; s2[t] = vobj; s3[t] = vcls;
    __syncthreads();
    for (int st = 128; st > 0; st >>= 1) {
        if (t < st) {
            s0[t] += s0[t + st]; s1[t] += s1[t + st];
            s2[t] += s2[t + st]; s3[t] += s3[t + st];
        }
        __syncthreads();
    }
    if (t == 0) {
        float* p = partials + (long)blockIdx.x * 4;
        p[0] = s0[0]; p[1] = s1[0]; p[2] = s2[0]; p[3] = s3[0];
    }
}

__global__ __launch_bounds__(256) void k_final(const float* __restrict__ partials,
                                               float* __restrict__ out) {
    __shared__ float s0[256], s1[256], s2[256], s3[256];
    int t = threadIdx.x;
    float a0 = 0.f, a1 = 0.f, a2 = 0.f, a3 = 0.f;
    for (int p = t; p < NPART; p += 256) {
        const float* q = partials + (long)p * 4;
        a0 += q[0]; a1 += q[1]; a2 += q[2]; a3 += q[3];
    }
    s0[t] = a0; s1[t] = a1; s2[t] = a2; s3[t] = a3;
    __syncthreads();
    for (int st = 128; st > 0; st >>= 1) {
        if (t < st) {
            s0[t] += s0[t + st]; s1[t] += s1[t + st];
            s2[t] += s2[t + st]; s3[t] += s3[t + st];
        }
        __syncthreads();
    }
    if (t == 0) {
        float nfg = fmaxf(s0[0], 1.f);
        out[0] = (5.f * s1[0] + s2[0] + s3[0]) / nfg;
    }
}

extern "C" void kernel_launch(void* const* d_in, const int* in_sizes, int n_in,
                              void* d_out, int out_size, void* d_ws, size_t ws_size,
                              hipStream_t stream) {
    (void)in_sizes; (void)n_in; (void)out_size; (void)ws_size;
    const float* pred   = (const float*)d_in[0];
    const float* obj    = (const float*)d_in[1];
    const float* cls    = (const float*)d_in[2];
    const float* labels = (const float*)d_in[3];
    const float* xs     = (const float*)d_in[4];
    const float* ys     = (const float*)d_in[5];
    const float* ss     = (const float*)d_in[6];

    char* ws = (char*)d_ws;
    size_t off = 0;
    float* sneg    = (float*)(ws + off); off += (size_t)BA * 4;
    float* fg0     = (float*)(ws + off); off += (size_t)BA * 4;
    int*   dynk    = (int*)  (ws + off); off += (size_t)BG * 4;
    int*   topk    = (int*)  (ws + off); off += (size_t)BG * 10 * 4;
    float* fg      = (float*)(ws + off); off += (size_t)BA * 4;
    int*   matched = (int*)  (ws + off); off += (size_t)BA * 4;
    float* prediou = (float*)(ws + off); off += (size_t)BA * 4;
    float* partials= (float*)(ws + off); off += (size_t)NPART * 4 * 4;

    k_sneg<<<BA / 128, 256, 0, stream>>>(cls, obj, sneg);
    k_fg0 <<<NPART,    256, 0, stream>>>(labels, xs, ys, ss, fg0);
    k_rows<<<BG,       256, 0, stream>>>(pred, obj, cls, labels, xs, ys, ss,
                                         sneg, fg0, dynk, topk);
    k_cols<<<NPART,    256, 0, stream>>>(pred, obj, cls, labels, xs, ys, ss,
                                         sneg, fg0, dynk, topk,
                                         fg, matched, prediou);
    k_loss<<<NPART,    256, 0, stream>>>(pred, obj, cls, labels,
                                         fg, matched, prediou, partials);
    k_final<<<1,       256, 0, stream>>>(partials, (float*)d_out);
}